// Point_contextual_entropy_parameter_6047313953103
// MI455X (gfx1250) — compile-verified
//
#include <hip/hip_runtime.h>

typedef __attribute__((ext_vector_type(2))) float v2f;
typedef __attribute__((ext_vector_type(8))) float v8f;

// ---------------------------------------------------------------------------
// Repack W [27, 64, COUT] into per-lane WMMA B-fragment order:
//   WP[((k*16 + c)*NT + t)*32 + lane] = { W[k][4c+2*hi][t*16+col],
//                                         W[k][4c+2*hi+1][t*16+col] }
// so the inner loop reads one coalesced b64 per fragment.
// ---------------------------------------------------------------------------
__global__ void repack_weights(const float* __restrict__ W,
                               v2f* __restrict__ WP, int cout) {
  const int nt = cout / 16;
  const int total = 27 * 16 * nt * 32;
  const int e = blockIdx.x * blockDim.x + threadIdx.x;
  if (e >= total) return;
  const int lane = e & 31;
  int rest = e >> 5;
  const int t = rest % nt; rest /= nt;
  const int c = rest % 16;
  const int k = rest / 16;
  const int hi  = lane >> 4;
  const int col = lane & 15;
  const int r = 4 * c + 2 * hi;
  v2f o;
  o.x = W[((size_t)k * 64 + r) * cout + t * 16 + col];
  o.y = W[((size_t)k * 64 + r + 1) * cout + t * 16 + col];
  WP[e] = o;
}

__global__ void init_zero_row(float* __restrict__ z) {
  z[threadIdx.x] = 0.0f;  // 64-float zero feature row for missing neighbors
}

// ---------------------------------------------------------------------------
// One wave handles MT M-tiles (MT*16 points). acc: MT x (COUT/16) x v8f.
// fp32 WMMA 16x16x4 => exact f32 numerics vs the reference.
// ---------------------------------------------------------------------------
template <int COUT, int MT, bool RELU>
__launch_bounds__(256)
__global__ void sparse_conv_wmma(const float* __restrict__ feats,  // [n, 64]
                                 const int*   __restrict__ nbr,    // [n, 27]
                                 const v2f*   __restrict__ WP,     // packed
                                 const float* __restrict__ bias,   // [COUT]
                                 const float* __restrict__ zrow,   // [64] zeros
                                 float*       __restrict__ out,    // [n, COUT]
                                 int n) {
  constexpr int NT = COUT / 16;

  const int lane  = threadIdx.x & 31;
  const int gwave = blockIdx.x * (blockDim.x >> 5) + (threadIdx.x >> 5);
  const int base  = gwave * (MT * 16);
  if (base >= n) return;  // wave-uniform exit; EXEC all-ones inside

  const int row = lane & 15;  // M row in tile / N col for B,C,D
  const int hi  = lane >> 4;  // K sub-pair select for A/B; M+8 for C/D

  v8f acc[MT][NT];
#pragma unroll
  for (int m = 0; m < MT; ++m)
#pragma unroll
    for (int t = 0; t < NT; ++t)
      acc[m][t] = (v8f){0.f, 0.f, 0.f, 0.f, 0.f, 0.f, 0.f, 0.f};

  // Per-M-tile neighbor-list offsets (clamped for a partial final wave).
  int poff[MT];
#pragma unroll
  for (int m = 0; m < MT; ++m)
    poff[m] = min(base + m * 16 + row, n - 1) * 27;

  for (int k = 0; k < 27; ++k) {
    // Resolve gather pointers; missing neighbors read the zero row (no
    // per-chunk cndmask needed).
    const float* ap[MT];
#pragma unroll
    for (int m = 0; m < MT; ++m) {
      const int i = nbr[poff[m] + k];
      ap[m] = (i >= 0) ? (feats + (size_t)i * 64) : zrow;
    }
    const v2f* wpk = WP + (size_t)k * 16 * NT * 32;

#pragma unroll
    for (int c = 0; c < 16; ++c) {       // 16 K-chunks of 4 cover Cin=64
      const int kb = c * 4 + hi * 2;     // A layout: lane hi picks K pair
      v2f a[MT];
#pragma unroll
      for (int m = 0; m < MT; ++m)
        a[m] = *(const v2f*)(ap[m] + kb);

#pragma unroll
      for (int t = 0; t < NT; ++t) {
        const v2f b = wpk[(c * NT + t) * 32 + lane];  // one coalesced b64
#pragma unroll
        for (int m = 0; m < MT; ++m)
          acc[m][t] = __builtin_amdgcn_wmma_f32_16x16x4_f32(
              false, a[m], false, b, (short)0, acc[m][t], false, false);
      }
    }
  }

  // Epilogue: bias (+ReLU); C/D layout: VGPR v -> row v + 8*hi, lane -> col.
#pragma unroll
  for (int t = 0; t < NT; ++t) {
    const float bv = bias[t * 16 + row];
#pragma unroll
    for (int m = 0; m < MT; ++m) {
#pragma unroll
      for (int v = 0; v < 8; ++v) {
        const int r = base + m * 16 + v + 8 * hi;
        float x = acc[m][t][v] + bv;
        if (RELU) x = fmaxf(x, 0.0f);
        if (r < n) out[(size_t)r * COUT + t * 16 + row] = x;
      }
    }
  }
}

extern "C" void kernel_launch(void* const* d_in, const int* in_sizes, int n_in,
                              void* d_out, int out_size, void* d_ws, size_t ws_size,
                              hipStream_t stream) {
  const float* feats = (const float*)d_in[0];
  const float* W1    = (const float*)d_in[1];
  const float* b1    = (const float*)d_in[2];
  const float* W2    = (const float*)d_in[3];
  const float* b2    = (const float*)d_in[4];
  const float* W3    = (const float*)d_in[5];
  const float* b3    = (const float*)d_in[6];
  const int*   nbr   = (const int*)d_in[7];

  const int n = in_sizes[0] / 64;  // N points (feats is [N, 64])

  // Workspace layout (floats)
  float* ws   = (float*)d_ws;
  float* h1   = ws;                         // [n, 64]
  float* h2   = h1 + (size_t)n * 64;        // [n, 64]
  float* wp1f = h2 + (size_t)n * 64;        // 27*16*4*32*2 = 110592 floats
  float* wp2f = wp1f + 27 * 16 * 4 * 32 * 2;
  float* wp3f = wp2f + 27 * 16 * 4 * 32 * 2;  // 27*16*8*32*2 = 221184 floats
  float* zrow = wp3f + 27 * 16 * 8 * 32 * 2;  // [64] zeros
  v2f* wp1 = (v2f*)wp1f;
  v2f* wp2 = (v2f*)wp2f;
  v2f* wp3 = (v2f*)wp3f;

  // --- Stage 0: repack weights into B-fragment order + zero row ---
  {
    const int tot64  = 27 * 16 * 4 * 32;  // COUT=64 packed elements
    const int tot128 = 27 * 16 * 8 * 32;  // COUT=128 packed elements
    repack_weights<<<(tot64 + 255) / 256, 256, 0, stream>>>(W1, wp1, 64);
    repack_weights<<<(tot64 + 255) / 256, 256, 0, stream>>>(W2, wp2, 64);
    repack_weights<<<(tot128 + 255) / 256, 256, 0, stream>>>(W3, wp3, 128);
    init_zero_row<<<1, 64, 0, stream>>>(zrow);
  }

  // --- 3 conv layers ---
  {
    // Layers 1-2: COUT=64, MT=4 (64 points/wave), 8 waves/block.
    const int waves64  = (n + 63) / 64;
    dim3 blk(256), grd64((waves64 + 7) / 8);
    sparse_conv_wmma<64, 4, true><<<grd64, blk, 0, stream>>>(
        feats, nbr, wp1, b1, zrow, h1, n);
    sparse_conv_wmma<64, 4, true><<<grd64, blk, 0, stream>>>(
        h1, nbr, wp2, b2, zrow, h2, n);

    // Layer 3: COUT=128, MT=2 (32 points/wave).
    const int waves128 = (n + 31) / 32;
    dim3 grd128((waves128 + 7) / 8);
    sparse_conv_wmma<128, 2, false><<<grd128, blk, 0, stream>>>(
        h2, nbr, wp3, b3, zrow, (float*)d_out, n);
  }
}